// ReversibleMultiScaleBlock_84765474553879
// MI455X (gfx1250) — compile-verified
//
#include <hip/hip_runtime.h>
#include <math.h>
#include <stdint.h>

// ---------------------------------------------------------------------------
// ReversibleMultiScaleBlock for MI455X (gfx1250, wave32, WMMA).
// All GEMM-like math runs on V_WMMA_F32_16X16X4_F32 (full fp32 precision).
// Global->LDS tile staging uses GLOBAL_LOAD_ASYNC_TO_LDS_B128 (ASYNCcnt).
// ---------------------------------------------------------------------------

typedef __attribute__((ext_vector_type(2))) float v2f;
typedef __attribute__((ext_vector_type(8))) float v8f;
typedef int v4i_vs __attribute__((vector_size(16)));  // builtin param type

#define T_TOK 25088      // B*H*W = 8*56*56
#define DIMC  384
#define NHEAD 8
#define HDC   48
#define HIDC  1536
#define EPSF  1e-5f
#define SCALEF 0.14433756729740643f   // 48^-0.5

#define AS1 __attribute__((address_space(1)))
#define AS3 __attribute__((address_space(3)))

// fp32 WMMA fragment layout (per CDNA5 ISA):
//  A 16x4 : lane L -> row m=L&15, khalf=L>>4; vgpr j holds K = 2*khalf + j
//  B 4x16 : lane L -> col n=L&15, khalf=L>>4; vgpr j holds K = 2*khalf + j
//  C/D    : vgpr r -> (m = r + 8*(L>>4), n = L&15)
static __device__ __forceinline__ v8f wmma_f32(v2f a, v2f b, v8f c) {
  return __builtin_amdgcn_wmma_f32_16x16x4_f32(false, a, false, b, (short)0, c,
                                               false, false);
}

// ---- async global->LDS 16B copy (per-lane addresses, tracked by ASYNCcnt) --
// Generic LDS pointer's low 32 bits are the LDS byte offset (FLAT aperture
// rule); integer-cast to an AS(3) pointer for builtin.
static __device__ __forceinline__ void cp16_async(void* l, const void* g) {
#if __has_builtin(__builtin_amdgcn_global_load_async_to_lds_b128)
  __builtin_amdgcn_global_load_async_to_lds_b128(
      (AS1 v4i_vs*)(uintptr_t)g,
      (AS3 v4i_vs*)(uint32_t)(uintptr_t)l, 0, 0);
#else
  asm volatile("global_load_async_to_lds_b128 %0, %1, off"
               :: "v"((uint32_t)(uintptr_t)l), "v"(g) : "memory");
#endif
}

static __device__ __forceinline__ void async_join() {
#if __has_builtin(__builtin_amdgcn_s_wait_asynccnt)
  __builtin_amdgcn_s_wait_asynccnt(0);
#else
  asm volatile("s_wait_asynccnt 0" ::: "memory");
#endif
}

// ------------------------------- LayerNorm ---------------------------------
__global__ void ln384_kernel(const float* __restrict__ x,
                             const float* __restrict__ w,
                             const float* __restrict__ b,
                             float* __restrict__ out) {
  const int wave = threadIdx.x >> 5, lane = threadIdx.x & 31;
  const int row = blockIdx.x * 8 + wave;
  const float* xr = x + (size_t)row * DIMC;
  float vals[12];
  float s = 0.f;
#pragma unroll
  for (int i = 0; i < 12; ++i) { vals[i] = xr[lane + 32 * i]; s += vals[i]; }
#pragma unroll
  for (int m = 16; m >= 1; m >>= 1) s += __shfl_xor(s, m, 32);
  const float mean = s * (1.f / DIMC);
  float ss = 0.f;
#pragma unroll
  for (int i = 0; i < 12; ++i) { float d = vals[i] - mean; ss += d * d; }
#pragma unroll
  for (int m = 16; m >= 1; m >>= 1) ss += __shfl_xor(ss, m, 32);
  const float inv = rsqrtf(ss * (1.f / DIMC) + EPSF);
  float* orow = out + (size_t)row * DIMC;
#pragma unroll
  for (int i = 0; i < 12; ++i) {
    int c = lane + 32 * i;
    orow[c] = (vals[i] - mean) * inv * w[c] + b[c];
  }
}

// ------------------------- WMMA GEMM (fp32) --------------------------------
// C[M,N] = A[M,K] @ W[K,N] + bias  (+GELU | +residual)
// BM=32, BN=64, BK=32; 256 threads = 8 waves in a 2x4 grid of 16x16 tiles.
// Requires M%32==0, N%64==0, K%32==0 (true for every GEMM in this block).
// LDS strides: As 36 (144B rows: 16B-aligned for async b128; period-16
// bank pattern -> conflict-free column reads), Bs 68 (row-major reads).
template <int EPI>  // 0: bias, 1: bias+gelu(exact), 2: bias+residual
__global__ void gemm_wmma(const float* __restrict__ A,
                          const float* __restrict__ W,
                          const float* __restrict__ bias,
                          const float* __restrict__ R,
                          float* __restrict__ C, int M, int N, int K) {
  __shared__ float As[32][36];
  __shared__ float Bs[32][68];
  const int t = threadIdx.x;
  const int wave = t >> 5, lane = t & 31, li = lane & 15, half = lane >> 4;
  const int m0 = blockIdx.x * 32, n0 = blockIdx.y * 64;
  const int wm = (wave & 1) * 16, wn = (wave >> 1) * 16;
  const int ar = t >> 3, ac4 = (t & 7) * 4;        // A: 32 rows x 8 float4
  const int br = t >> 4, bc4 = (t & 15) * 4;       // B: 16 rows x 16 float4 x2

  v8f acc = {};
  for (int k0 = 0; k0 < K; k0 += 32) {
    cp16_async(&As[ar][ac4], A + (size_t)(m0 + ar) * K + k0 + ac4);
    cp16_async(&Bs[br][bc4], W + (size_t)(k0 + br) * N + n0 + bc4);
    cp16_async(&Bs[br + 16][bc4], W + (size_t)(k0 + br + 16) * N + n0 + bc4);
    async_join();
    __syncthreads();
#pragma unroll
    for (int kk = 0; kk < 32; kk += 4) {
      const int k2 = kk + 2 * half;
      v2f a, bf;
      a.x = As[wm + li][k2];     a.y = As[wm + li][k2 + 1];
      bf.x = Bs[k2][wn + li];    bf.y = Bs[k2 + 1][wn + li];
      acc = wmma_f32(a, bf, acc);
    }
    __syncthreads();
  }
  const int col = n0 + wn + li;
  const float bv = bias[col];
#pragma unroll
  for (int r = 0; r < 8; ++r) {
    const int row = m0 + wm + r + 8 * half;
    float v = acc[r] + bv;
    if (EPI == 1) v = 0.5f * v * (1.0f + erff(v * 0.7071067811865475f));
    if (EPI == 2) v += R[(size_t)row * N + col];
    C[(size_t)row * N + col] = v;
  }
}

// ----------------- depthwise 3x3 conv + LayerNorm(HD=48) -------------------
// Input: qkv [B,H,W,3,NH,HD]; output: [B*NH, H, W, HD] with conv+LN fused.
// 16 lanes per pixel, 3 channels per lane; half-wave shfl reduction for LN.
__global__ void dwconv_ln(const float* __restrict__ qkv, int which,
                          const float* __restrict__ pw,
                          const float* __restrict__ nw,
                          const float* __restrict__ nb,
                          float* __restrict__ out) {
  const int t = threadIdx.x;
  const int wave = t >> 5, lane = t & 31, half = lane >> 4, li = lane & 15;
  const int p = blockIdx.x * 16 + wave * 2 + half;   // pixel id in [0, 64*3136)
  const int n = p / 3136, rem = p - n * 3136;
  const int y = rem / 56, x = rem - y * 56;
  const int b = n >> 3, head = n & 7;
  const int base_c = which * DIMC + head * HDC;

  float a[3];
#pragma unroll
  for (int j = 0; j < 3; ++j) {
    const int c = li * 3 + j;
    float s = 0.f;
#pragma unroll
    for (int dy = -1; dy <= 1; ++dy) {
      const int yy = y + dy;
      if (yy < 0 || yy >= 56) continue;
#pragma unroll
      for (int dx = -1; dx <= 1; ++dx) {
        const int xx = x + dx;
        if (xx < 0 || xx >= 56) continue;
        s += pw[((dy + 1) * 3 + (dx + 1)) * HDC + c] *
             qkv[(size_t)((b * 56 + yy) * 56 + xx) * 1152 + base_c + c];
      }
    }
    a[j] = s;
  }
  float s = a[0] + a[1] + a[2];
  float ss = a[0] * a[0] + a[1] * a[1] + a[2] * a[2];
#pragma unroll
  for (int m = 8; m >= 1; m >>= 1) {
    s += __shfl_xor(s, m, 32);
    ss += __shfl_xor(ss, m, 32);
  }
  const float mean = s * (1.f / 48.f);
  const float var = ss * (1.f / 48.f) - mean * mean;
  const float inv = rsqrtf(var + EPSF);
  const size_t ob = (size_t)(n * 3136 + y * 56 + x) * HDC;
#pragma unroll
  for (int j = 0; j < 3; ++j) {
    const int c = li * 3 + j;
    out[ob + c] = (a[j] - mean) * inv * nw[c] + nb[c];
  }
}

// ---------------------- fused window attention -----------------------------
// One workgroup per window (1024 windows of 196 tokens x 48 dims).
// Entire attention (scores + decomposed rel-pos bias + softmax + AV + ori_q
// residual) runs out of ~316KB LDS (CDNA5 WGP has 320KB). Output is written
// directly in the permuted [B,H,W,DIM] layout consumed by the proj GEMM.
#define CSTR 52    // q/k/v LDS row stride: 208B rows (async-aligned),
                   // 52*i mod 64 has period 16 -> conflict-free column reads
#define SSTR 209   // score LDS row stride (odd -> conflict-free)
#define ATT_SMEM_FLOATS (2 * 208 * CSTR + 196 * CSTR + 208 * SSTR + 2 * 196 * 14)

__global__ void win_attn(const float* __restrict__ qg,
                         const float* __restrict__ kg,
                         const float* __restrict__ vg,
                         const float* __restrict__ relh,
                         const float* __restrict__ relw,
                         float* __restrict__ outp) {
  extern __shared__ float smem[];
  float* qs = smem;                  // [208][CSTR] (rows 196.. zero)
  float* ks = qs + 208 * CSTR;       // [208][CSTR] (rows 196.. zero)
  float* vs = ks + 208 * CSTR;       // [196][CSTR]
  float* S  = vs + 196 * CSTR;       // [208][SSTR]
  float* Bh = S + 208 * SSTR;        // [196][14]
  float* Bw = Bh + 196 * 14;         // [196][14]

  const int t = threadIdx.x;
  const int wave = t >> 5, lane = t & 31, li = lane & 15, half = lane >> 4;
  const int wid = blockIdx.x;
  const int n = wid >> 4, wi = wid & 15;
  const int wy = (wi >> 2) * 14, wx = (wi & 3) * 14;
  const int b = n >> 3, head = n & 7;

  // async-load q,k,v window tiles (196 rows x 12 float4 each)
  for (int idx = t; idx < 196 * 12; idx += 256) {
    const int row = idx / 12, c4 = (idx - row * 12) * 4;
    const int y = wy + row / 14, x = wx + row % 14;
    const size_t g = (size_t)((n * 56 + y) * 56 + x) * HDC + c4;
    cp16_async(qs + row * CSTR + c4, qg + g);
    cp16_async(ks + row * CSTR + c4, kg + g);
    cp16_async(vs + row * CSTR + c4, vg + g);
  }
  for (int idx = t; idx < 12 * HDC; idx += 256) {   // zero M/N pad rows
    const int row = 196 + idx / HDC, c = idx % HDC;
    qs[row * CSTR + c] = 0.f;
    ks[row * CSTR + c] = 0.f;
  }
  async_join();
  __syncthreads();

  // decomposed rel-pos bias tables: Bh[q][kh] = q . rel_h[h(q)-kh+13]
  for (int idx = t; idx < 2 * 196 * 14; idx += 256) {
    const int sel = idx >= 196 * 14;
    const int id2 = sel ? idx - 196 * 14 : idx;
    const int qi = id2 / 14, kj = id2 - (id2 / 14) * 14;
    const int coord = sel ? (qi % 14) : (qi / 14);
    const float* rp = (sel ? relw : relh) + (size_t)(coord - kj + 13) * HDC;
    const float* qp = qs + qi * CSTR;
    float s = 0.f;
#pragma unroll
    for (int c = 0; c < HDC; ++c) s += qp[c] * rp[c];
    (sel ? Bw : Bh)[id2] = s;
  }
  __syncthreads();

  // scores: S = SCALE * q k^T + bias   (13x13 tiles of 16x16, K=48)
  for (int ti = wave; ti < 169; ti += 8) {
    const int tm = (ti / 13) * 16, tn = (ti % 13) * 16;
    v8f acc = {};
#pragma unroll
    for (int kk = 0; kk < HDC; kk += 4) {
      const int k2 = kk + 2 * half;
      v2f a, bf;
      a.x = qs[(tm + li) * CSTR + k2];  a.y = qs[(tm + li) * CSTR + k2 + 1];
      bf.x = ks[(tn + li) * CSTR + k2]; bf.y = ks[(tn + li) * CSTR + k2 + 1];
      acc = wmma_f32(a, bf, acc);
    }
    const int col = tn + li;
#pragma unroll
    for (int r = 0; r < 8; ++r) {
      const int row = tm + r + 8 * half;
      const int rr = row < 196 ? row : 0, cc = col < 196 ? col : 0;
      const float bias = Bh[rr * 14 + cc / 14] + Bw[rr * 14 + cc % 14];
      S[row * SSTR + col] = SCALEF * acc[r] + bias;
    }
  }
  __syncthreads();

  // softmax over 196 keys, one wave per row
  for (int row = wave; row < 196; row += 8) {
    float* Sr = S + row * SSTR;
    float mx = -1e30f;
    for (int c = lane; c < 196; c += 32) mx = fmaxf(mx, Sr[c]);
#pragma unroll
    for (int m = 16; m >= 1; m >>= 1) mx = fmaxf(mx, __shfl_xor(mx, m, 32));
    float sum = 0.f;
    for (int c = lane; c < 196; c += 32) {
      const float e = expf(Sr[c] - mx);
      Sr[c] = e;
      sum += e;
    }
#pragma unroll
    for (int m = 16; m >= 1; m >>= 1) sum += __shfl_xor(sum, m, 32);
    const float inv = 1.f / sum;
    for (int c = lane; c < 196; c += 32) Sr[c] *= inv;
  }
  __syncthreads();

  // o = attn @ v + ori_q, written in permuted [B,H,W,DIM] layout
  for (int ti = wave; ti < 39; ti += 8) {
    const int tm = (ti / 3) * 16, tn = (ti % 3) * 16;
    v8f acc = {};
    for (int kk = 0; kk < 196; kk += 4) {   // K=196 exactly (49 steps)
      const int k2 = kk + 2 * half;
      v2f a, bf;
      a.x = S[(tm + li) * SSTR + k2];  a.y = S[(tm + li) * SSTR + k2 + 1];
      bf.x = vs[k2 * CSTR + tn + li];  bf.y = vs[(k2 + 1) * CSTR + tn + li];
      acc = wmma_f32(a, bf, acc);
    }
    const int c = tn + li;
#pragma unroll
    for (int r = 0; r < 8; ++r) {
      const int row = tm + r + 8 * half;
      if (row < 196) {
        const int y = wy + row / 14, x = wx + row % 14;
        outp[(size_t)((b * 56 + y) * 56 + x) * DIMC + head * HDC + c] =
            acc[r] + qs[row * CSTR + c];
      }
    }
  }
}

// ------------------------------- launch ------------------------------------
extern "C" void kernel_launch(void* const* d_in, const int* in_sizes, int n_in,
                              void* d_out, int out_size, void* d_ws,
                              size_t ws_size, hipStream_t stream) {
  (void)in_sizes; (void)n_in; (void)out_size; (void)ws_size;
  const float* x1 = (const float*)d_in[0];
  const float* x2 = (const float*)d_in[1];
  const float* n1w = (const float*)d_in[2];
  const float* n1b = (const float*)d_in[3];
  const float* qkv_w = (const float*)d_in[4];
  const float* qkv_b = (const float*)d_in[5];
  const float* pqw = (const float*)d_in[6];
  const float* pkw = (const float*)d_in[7];
  const float* pvw = (const float*)d_in[8];
  const float* nqw = (const float*)d_in[9];
  const float* nqb = (const float*)d_in[10];
  const float* nkw = (const float*)d_in[11];
  const float* nkb = (const float*)d_in[12];
  const float* nvw = (const float*)d_in[13];
  const float* nvb = (const float*)d_in[14];
  const float* relh = (const float*)d_in[15];
  const float* relw = (const float*)d_in[16];
  const float* prw = (const float*)d_in[17];
  const float* prb = (const float*)d_in[18];
  const float* n2w = (const float*)d_in[19];
  const float* n2b = (const float*)d_in[20];
  const float* f1w = (const float*)d_in[21];
  const float* f1b = (const float*)d_in[22];
  const float* f2w = (const float*)d_in[23];
  const float* f2b = (const float*)d_in[24];

  // workspace layout (floats), aggressively reused; peak 67,436,544 f (~270MB)
  float* ws = (float*)d_ws;
  const size_t TD = (size_t)T_TOK * DIMC;          // 9,633,792
  float* xn = ws;                                  // [0, TD)
  float* qkv = ws + TD;                            // [TD, 4TD)
  float* qb = ws + 4 * TD;                         // [4TD, 5TD)
  float* kb = ws + 5 * TD;                         // [5TD, 6TD)
  float* vb = ws + 6 * TD;                         // [6TD, 7TD)
  float* projin = ws;                              // reuse xn (dead)
  float* yn = ws + TD;                             // reuse qkv (dead)
  float* hidden = ws + 2 * TD;                     // reuse qkv tail + q/k (dead)

  float* y1 = (float*)d_out;
  float* y2 = y1 + TD;

  // 1. xn = LN(x1)
  ln384_kernel<<<T_TOK / 8, 256, 0, stream>>>(x1, n1w, n1b, xn);
  // 2. qkv = xn @ qkv_w + qkv_b
  gemm_wmma<0><<<dim3(T_TOK / 32, 1152 / 64), 256, 0, stream>>>(
      xn, qkv_w, qkv_b, nullptr, qkv, T_TOK, 1152, DIMC);
  // 3. q/k/v = LN(dwconv3x3(qkv[.]))
  dwconv_ln<<<12544, 256, 0, stream>>>(qkv, 0, pqw, nqw, nqb, qb);
  dwconv_ln<<<12544, 256, 0, stream>>>(qkv, 1, pkw, nkw, nkb, kb);
  dwconv_ln<<<12544, 256, 0, stream>>>(qkv, 2, pvw, nvw, nvb, vb);
  // 4. fused window attention -> projin (permuted layout), residual +ori_q
  const size_t att_smem = (size_t)ATT_SMEM_FLOATS * sizeof(float);  // 323,136B
  (void)hipFuncSetAttribute((const void*)win_attn,
                            hipFuncAttributeMaxDynamicSharedMemorySize,
                            (int)att_smem);
  win_attn<<<1024, 256, att_smem, stream>>>(qb, kb, vb, relh, relw, projin);
  // 5. y1 = x1 + projin @ proj_w + proj_b
  gemm_wmma<2><<<dim3(T_TOK / 32, DIMC / 64), 256, 0, stream>>>(
      projin, prw, prb, x1, y1, T_TOK, DIMC, DIMC);
  // 6. yn = LN(y1)
  ln384_kernel<<<T_TOK / 8, 256, 0, stream>>>(y1, n2w, n2b, yn);
  // 7. hidden = gelu(yn @ fc1_w + fc1_b)
  gemm_wmma<1><<<dim3(T_TOK / 32, HIDC / 64), 256, 0, stream>>>(
      yn, f1w, f1b, nullptr, hidden, T_TOK, HIDC, DIMC);
  // 8. y2 = x2 + hidden @ fc2_w + fc2_b
  gemm_wmma<2><<<dim3(T_TOK / 32, DIMC / 64), 256, 0, stream>>>(
      hidden, f2w, f2b, x2, y2, T_TOK, DIMC, HIDC);
}